// Aggregator_10445360464162
// MI455X (gfx1250) — compile-verified
//
#include <hip/hip_runtime.h>

typedef float v2f __attribute__((ext_vector_type(2)));
typedef float v8f __attribute__((ext_vector_type(8)));

#define D 256
#define KSTEPS (D / 4)        // 64 K-steps of 4
#define NEG_SLOPE 0.01f
#define ROWS 32               // M rows per block = 2 WMMA M-tiles

// ---------------- zero the side accumulator (lives in d_out) ----------------
__global__ __launch_bounds__(256) void zero_kernel(float4* __restrict__ p, int n4) {
    int i = blockIdx.x * 256 + threadIdx.x;
    if (i < n4) p[i] = make_float4(0.f, 0.f, 0.f, 0.f);
}

// ---------------- COO SpMM scatter: side[row] += val * ego[col] -------------
// One wave per edge: 32 lanes x 8 floats, b128 gathers, 8 fp32 atomics/lane.
__global__ __launch_bounds__(256) void scatter_kernel(
    const float* __restrict__ vals, const int* __restrict__ rows,
    const int* __restrict__ cols, const float* __restrict__ ego,
    float* __restrict__ side, int E) {
    unsigned tid  = blockIdx.x * 256u + threadIdx.x;
    unsigned edge = tid >> 5;
    unsigned lane = tid & 31u;
    if (edge >= (unsigned)E) return;
    int   r = rows[edge];
    int   c = cols[edge];
    float v = vals[edge];
    const float4* src = (const float4*)(ego + (size_t)c * D + lane * 8);
    float4 e0 = src[0];
    float4 e1 = src[1];
    float* dst = side + (size_t)r * D + lane * 8;
    atomicAdd(dst + 0, v * e0.x);
    atomicAdd(dst + 1, v * e0.y);
    atomicAdd(dst + 2, v * e0.z);
    atomicAdd(dst + 3, v * e0.w);
    atomicAdd(dst + 4, v * e1.x);
    atomicAdd(dst + 5, v * e1.y);
    atomicAdd(dst + 6, v * e1.z);
    atomicAdd(dst + 7, v * e1.w);
}

// ---------------- pack W into B-fragment-ready layout -----------------------
// P[ks*1024 + ntIdx*64 + lane*2 + j] = W[(4*ks + (lane>>4)*2 + j)*D + ntIdx*16 + (lane&15)]
// -> consumer loads one coalesced b64 per fragment.
__global__ __launch_bounds__(256) void pack_w_kernel(const float* __restrict__ W,
                                                     float* __restrict__ P) {
    int t = blockIdx.x * 256 + threadIdx.x;      // KSTEPS*16*32 = 32768
    if (t >= KSTEPS * 16 * 32) return;
    int lane  = t & 31;
    int ntIdx = (t >> 5) & 15;
    int ks    = t >> 9;
    int col   = ntIdx * 16 + (lane & 15);
    int krow  = 4 * ks + ((lane >> 4) << 1);
    v2f v;
    v.x = W[(size_t)krow * D + col];
    v.y = W[(size_t)(krow + 1) * D + col];
    *(v2f*)(P + (size_t)t * 2) = v;
}

__device__ __forceinline__ v8f wmma4(v2f a, v2f b, v8f c) {
    return __builtin_amdgcn_wmma_f32_16x16x4_f32(false, a, false, b, (short)0, c,
                                                 false, false);
}

// ---------------- fused bi-interaction: out = lrelu(x1@W1+b1)+lrelu(x2@W2+b2)
// Block = 32-row M-tile (2 WMMA tiles). 8 waves x 2 N-tiles cover N=256.
// Per K-step per wave: 8 wmma, 4 B-frag loads (b64 packed), 4 A-frag ds b64.
// LDS uses XOR swizzle addr = row*256 + (k ^ ((row&15)<<2)) -> 64KB, no pad,
// conflict-free b64 A reads.
template <bool PACKED>
__global__ __launch_bounds__(256) void fused_gemm_kernel(
    const float* __restrict__ ego,  const float* __restrict__ side,
    const float* __restrict__ W1,   const float* __restrict__ b1,
    const float* __restrict__ W2,   const float* __restrict__ b2,
    const float* __restrict__ P1,   const float* __restrict__ P2,
    float* __restrict__ out, int N) {
    __shared__ float x1T[ROWS * D];
    __shared__ float x2T[ROWS * D];

    const int m0 = blockIdx.x * ROWS;

    // Stage x1 = ego + side, x2 = ego * side (swizzled) for this 32-row tile.
    for (int idx = threadIdx.x; idx < ROWS * (D / 4); idx += 256) {
        int row = idx >> 6;
        int c4  = (idx & 63) << 2;
        int sc  = c4 ^ ((row & 15) << 2);      // float4 block stays contiguous
        float4 e = make_float4(0.f, 0.f, 0.f, 0.f);
        float4 s = make_float4(0.f, 0.f, 0.f, 0.f);
        if (m0 + row < N) {
            size_t g = (size_t)(m0 + row) * D + c4;
            e = *(const float4*)(ego  + g);
            s = *(const float4*)(side + g);
        }
        float* p1 = x1T + row * D + sc;
        float* p2 = x2T + row * D + sc;
        p1[0] = e.x + s.x; p1[1] = e.y + s.y; p1[2] = e.z + s.z; p1[3] = e.w + s.w;
        p2[0] = e.x * s.x; p2[1] = e.y * s.y; p2[2] = e.z * s.z; p2[3] = e.w * s.w;
    }
    __syncthreads();

    const int lane = threadIdx.x & 31;
    const int wave = threadIdx.x >> 5;
    const int half = lane >> 4;
    const int mrow = lane & 15;
    const int koff = half << 1;
    const int nA   = wave * 32;
    const int nB   = nA + 16;
    const int sw   = mrow << 2;                // XOR swizzle, same for row & row+16

    v8f acc[2][2][2] = {};                     // [mtile][nsel][path]

    const float* x1r0 = x1T + mrow * D;
    const float* x1r1 = x1T + (16 + mrow) * D;
    const float* x2r0 = x2T + mrow * D;
    const float* x2r1 = x2T + (16 + mrow) * D;

    #pragma unroll 4
    for (int ks = 0; ks < KSTEPS; ++ks) {
        int kk = ks * 4;
        int kc = (kk + koff) ^ sw;             // swizzled column of K pair
        v2f a1m0 = *(const v2f*)(x1r0 + kc);
        v2f a2m0 = *(const v2f*)(x2r0 + kc);
        v2f a1m1 = *(const v2f*)(x1r1 + kc);
        v2f a2m1 = *(const v2f*)(x2r1 + kc);

        v2f bW1A, bW1B, bW2A, bW2B;
        if (PACKED) {
            int pb = kk * 256 + lane * 2;      // ks*1024 + lane*2
            bW1A = *(const v2f*)(P1 + pb + (wave * 2)     * 64);
            bW1B = *(const v2f*)(P1 + pb + (wave * 2 + 1) * 64);
            bW2A = *(const v2f*)(P2 + pb + (wave * 2)     * 64);
            bW2B = *(const v2f*)(P2 + pb + (wave * 2 + 1) * 64);
        } else {
            int wbase = (kk + koff) * D + mrow;
            bW1A.x = W1[wbase + nA];  bW1A.y = W1[wbase + D + nA];
            bW1B.x = W1[wbase + nB];  bW1B.y = W1[wbase + D + nB];
            bW2A.x = W2[wbase + nA];  bW2A.y = W2[wbase + D + nA];
            bW2B.x = W2[wbase + nB];  bW2B.y = W2[wbase + D + nB];
        }

        acc[0][0][0] = wmma4(a1m0, bW1A, acc[0][0][0]);
        acc[0][0][1] = wmma4(a2m0, bW2A, acc[0][0][1]);
        acc[0][1][0] = wmma4(a1m0, bW1B, acc[0][1][0]);
        acc[0][1][1] = wmma4(a2m0, bW2B, acc[0][1][1]);
        acc[1][0][0] = wmma4(a1m1, bW1A, acc[1][0][0]);
        acc[1][0][1] = wmma4(a2m1, bW2A, acc[1][0][1]);
        acc[1][1][0] = wmma4(a1m1, bW1B, acc[1][1][0]);
        acc[1][1][1] = wmma4(a2m1, bW2B, acc[1][1][1]);
    }

    // Epilogue: bias + leaky-relu per path, sum, store.
    // C layout: VGPR i -> M = i + 8*half, N = lane&15.
    const float bs1A = b1[nA + mrow], bs2A = b2[nA + mrow];
    const float bs1B = b1[nB + mrow], bs2B = b2[nB + mrow];
    #pragma unroll
    for (int mt = 0; mt < 2; ++mt) {
        #pragma unroll
        for (int i = 0; i < 8; ++i) {
            int m = m0 + mt * 16 + i + (half << 3);
            if (m >= N) continue;
            float r1 = acc[mt][0][0][i] + bs1A; r1 = r1 > 0.f ? r1 : NEG_SLOPE * r1;
            float r2 = acc[mt][0][1][i] + bs2A; r2 = r2 > 0.f ? r2 : NEG_SLOPE * r2;
            out[(size_t)m * D + nA + mrow] = r1 + r2;
            float r3 = acc[mt][1][0][i] + bs1B; r3 = r3 > 0.f ? r3 : NEG_SLOPE * r3;
            float r4 = acc[mt][1][1][i] + bs2B; r4 = r4 > 0.f ? r4 : NEG_SLOPE * r4;
            out[(size_t)m * D + nB + mrow] = r3 + r4;
        }
    }
}

extern "C" void kernel_launch(void* const* d_in, const int* in_sizes, int n_in,
                              void* d_out, int out_size, void* d_ws, size_t ws_size,
                              hipStream_t stream) {
    const float* ego  = (const float*)d_in[0];
    const float* vals = (const float*)d_in[1];
    const float* W1   = (const float*)d_in[2];
    const float* b1   = (const float*)d_in[3];
    const float* W2   = (const float*)d_in[4];
    const float* b2   = (const float*)d_in[5];
    const int*   rows = (const int*)d_in[6];
    const int*   cols = (const int*)d_in[7];
    float* out = (float*)d_out;

    const int N = in_sizes[0] / D;   // 100000
    const int E = in_sizes[1];       // 1600000

    // side accumulator lives in d_out; fused kernel overwrites it in place
    // (each block reads exactly the rows it later writes).
    float* side = out;

    int n4 = (N * D) / 4;
    zero_kernel<<<(n4 + 255) / 256, 256, 0, stream>>>((float4*)side, n4);

    unsigned total = (unsigned)E * 32u;
    scatter_kernel<<<(total + 255u) / 256u, 256, 0, stream>>>(vals, rows, cols, ego,
                                                              side, E);

    const size_t packElems = (size_t)KSTEPS * 1024;          // per matrix
    const size_t packBytes = 2 * packElems * sizeof(float);  // 512 KB
    int gridM = (N + ROWS - 1) / ROWS;

    if (d_ws != nullptr && ws_size >= packBytes) {
        float* P1 = (float*)d_ws;
        float* P2 = P1 + packElems;
        int pt = KSTEPS * 16 * 32;
        pack_w_kernel<<<(pt + 255) / 256, 256, 0, stream>>>(W1, P1);
        pack_w_kernel<<<(pt + 255) / 256, 256, 0, stream>>>(W2, P2);
        fused_gemm_kernel<true><<<gridM, 256, 0, stream>>>(ego, side, W1, b1, W2, b2,
                                                           P1, P2, out, N);
    } else {
        fused_gemm_kernel<false><<<gridM, 256, 0, stream>>>(ego, side, W1, b1, W2, b2,
                                                            W1, W2, out, N);
    }
}